// FGCN_4879082848463
// MI455X (gfx1250) — compile-verified
//
#include <hip/hip_runtime.h>

// FGCN graph-Laplacian block for MI455X (gfx1250, wave32).
// All GEMMs use V_WMMA_F32_16X16X4_F32 to keep exact fp32 semantics.
// Fusion: out = x + relu(X@Zt - P@U + zb), U = (rou*t)@Zt  (LX never materialized).
// Round 2: all staging loops vectorized to float4 (global_load_b128 / ds b128).

#define BATCH 8
#define CH    256
#define NPIX  9216      // 96*96
#define WCH   64

typedef float v2f __attribute__((ext_vector_type(2)));
typedef float v8f __attribute__((ext_vector_type(8)));

static __device__ __forceinline__ v8f wmma4(v2f a, v2f b, v8f c) {
  // D = A(16x4 f32) x B(4x16 f32) + C(16x16 f32)
  return __builtin_amdgcn_wmma_f32_16x16x4_f32(false, a, false, b, (short)0, c, false, false);
}

// ---------------- workspace layout (floats) ----------------
static constexpr size_t OFF_PHI   = 0;                                        // [B][N][64]
static constexpr size_t OFF_DINV  = OFF_PHI   + (size_t)BATCH*NPIX*WCH;       // [B][N]
static constexpr size_t OFF_SPART = OFF_DINV  + (size_t)BATCH*NPIX;           // [B][72][64]
static constexpr size_t OFF_RS    = OFF_SPART + (size_t)BATCH*72*WCH;         // [B][64]
static constexpr size_t OFF_XMEAN = OFF_RS    + (size_t)BATCH*WCH;            // [B][256]
static constexpr size_t OFF_ROU   = OFF_XMEAN + (size_t)BATCH*CH;             // [B][64]
static constexpr size_t OFF_TPART = OFF_ROU   + (size_t)BATCH*WCH;            // [B][12][64][256]
static constexpr size_t OFF_T     = OFF_TPART + (size_t)BATCH*12*WCH*CH;      // [B][64][256]
static constexpr size_t OFF_U     = OFF_T     + (size_t)BATCH*WCH*CH;         // [B][64][256] (negated)
static constexpr size_t OFF_ZT    = OFF_U     + (size_t)BATCH*WCH*CH;         // [256][256]
// total ~6.73M floats (~25.7 MiB)

// ---------------- per-channel mean of x ----------------
__global__ __launch_bounds__(256) void k_colmean(const float* __restrict__ x,
                                                 float* __restrict__ xmean) {
  int c = blockIdx.x, b = blockIdx.y, tid = threadIdx.x;
  const float4* p = (const float4*)(x + ((size_t)b*CH + c)*NPIX);
  float s = 0.f;
  for (int i = tid; i < NPIX/4; i += 256) {        // 9 iterations of b128
    float4 v = p[i];
    s += v.x + v.y + v.z + v.w;
  }
  __shared__ float red[256];
  red[tid] = s; __syncthreads();
  for (int off = 128; off > 0; off >>= 1) {
    if (tid < off) red[tid] += red[tid + off];
    __syncthreads();
  }
  if (tid == 0) xmean[b*CH + c] = red[0] * (1.0f/(float)NPIX);
}

// ---------------- zt = zeta_w^T ----------------
__global__ void k_zt(const float* __restrict__ zw, float* __restrict__ zt) {
  int k = blockIdx.x, c = threadIdx.x;
  zt[k*CH + c] = zw[c*CH + k];
}

// ---------------- rou = sigmoid(xmean @ rou_w^T + rou_b) ----------------
__global__ __launch_bounds__(256) void k_rou(const float* __restrict__ xmean,
                                             const float* __restrict__ rw,
                                             const float* __restrict__ rb,
                                             float* __restrict__ rou) {
  int gid = blockIdx.x*256 + threadIdx.x;
  if (gid >= BATCH*WCH) return;
  int b = gid >> 6, w = gid & 63;
  float s = rb[w];
  const float4* xm = (const float4*)(xmean + b*CH);
  const float4* wr = (const float4*)(rw + w*CH);
  for (int c = 0; c < CH/4; ++c) {
    float4 a = xm[c], q = wr[c];
    s += a.x*q.x + a.y*q.y + a.z*q.z + a.w*q.w;
  }
  rou[gid] = 1.0f/(1.0f + expf(-s));
}

// ---------------- phi = relu(phi_w @ x + phi_b), stored [B][N][64]; + s partials ----------------
__global__ __launch_bounds__(256) void k_phi(const float* __restrict__ x,
                                             const float* __restrict__ pw,
                                             const float* __restrict__ pb,
                                             float* __restrict__ phi,
                                             float* __restrict__ spart) {
  __shared__ float lpw[WCH*68];     // [w][kk] pad 68 -> float4-aligned, banks 4w+kk
  __shared__ float lpart[16*WCH];   // per (wave,h) s-partials
  const int tid  = threadIdx.x;
  const int wave = tid >> 5, lane = tid & 31, h = lane >> 4, l = lane & 15;
  const int b  = blockIdx.y;
  const int n0 = blockIdx.x * 128;
  const int nrow = n0 + wave*16 + l;                 // A-matrix row (M = n)
  const float* xb = x + (size_t)b*CH*NPIX;
  v8f acc[4] = {};
  for (int kc = 0; kc < CH; kc += 64) {
    __syncthreads();
    for (int idx = tid; idx < WCH*16; idx += 256) {  // 64w x 16 float4
      int w = idx >> 4, kq = (idx & 15) * 4;
      float4 v = *(const float4*)(pw + w*CH + kc + kq);
      *(float4*)(lpw + w*68 + kq) = v;
    }
    __syncthreads();
    for (int ks = 0; ks < 64; ks += 4) {
      int kk = ks + 2*h;
      v2f a;                                         // A[m=n][k=c], coalesced over n
      a.x = xb[(size_t)(kc+kk  )*NPIX + nrow];
      a.y = xb[(size_t)(kc+kk+1)*NPIX + nrow];
      #pragma unroll
      for (int wt = 0; wt < 4; ++wt) {
        int w = wt*16 + l;
        v2f bb; bb.x = lpw[w*68 + kk]; bb.y = lpw[w*68 + kk + 1];
        acc[wt] = wmma4(a, bb, acc[wt]);
      }
    }
  }
  #pragma unroll
  for (int wt = 0; wt < 4; ++wt) {
    int w = wt*16 + l;
    float bias = pb[w];
    float part = 0.f;
    #pragma unroll
    for (int r = 0; r < 8; ++r) {
      int n = n0 + wave*16 + r + 8*h;                // D: M = r + 8h
      float v = fmaxf(acc[wt][r] + bias, 0.0f);
      phi[(((size_t)b*NPIX + n) << 6) + w] = v;
      part += v;
    }
    lpart[(wave*2 + h)*WCH + w] = part;              // unique slot per (wave,h,w)
  }
  __syncthreads();
  if (tid < WCH) {                                   // deterministic fixed-order sum
    float s = 0.f;
    for (int i = 0; i < 16; ++i) s += lpart[i*WCH + tid];
    spart[((size_t)b*72 + blockIdx.x)*WCH + tid] = s;
  }
}

// ---------------- rs = rou * sum_blocks(spart) ----------------
__global__ void k_rs(const float* __restrict__ spart, const float* __restrict__ rou,
                     float* __restrict__ rs) {
  int gid = blockIdx.x*256 + threadIdx.x;
  if (gid >= BATCH*WCH) return;
  int b = gid >> 6, w = gid & 63;
  float s = 0.f;
  for (int j = 0; j < 72; ++j) s += spart[((size_t)b*72 + j)*WCH + w];
  rs[gid] = rou[gid] * s;
}

// ---------------- dinv = rsqrt(phi . rs + 1e-8) ----------------
__global__ __launch_bounds__(256) void k_dinv(const float* __restrict__ phi,
                                              const float* __restrict__ rs,
                                              float* __restrict__ dinv) {
  __shared__ float lrs[WCH];
  int b = blockIdx.y;
  int n = blockIdx.x*256 + threadIdx.x;
  if (threadIdx.x < WCH) lrs[threadIdx.x] = rs[b*WCH + threadIdx.x];
  __syncthreads();
  const float4* pp = (const float4*)(phi + (((size_t)b*NPIX + n) << 6));
  float d = 0.f;
  #pragma unroll
  for (int w = 0; w < WCH/4; ++w) {                  // 16 b128 loads
    float4 v = pp[w];
    d += v.x*lrs[w*4] + v.y*lrs[w*4+1] + v.z*lrs[w*4+2] + v.w*lrs[w*4+3];
  }
  dinv[(size_t)b*NPIX + n] = rsqrtf(d + 1e-8f);
}

// ---------------- t partials: t = P^T @ X, split-K over 12 chunks of 768 rows ----------------
__global__ __launch_bounds__(256) void k_t(const float* __restrict__ x,
                                           const float* __restrict__ phi,
                                           const float* __restrict__ dinv,
                                           float* __restrict__ tpart) {
  __shared__ float Xs[16*264];   // B staging: [k=n_local][c], pad 264 (264%4==0)
  __shared__ float Ps[16*72];    // A staging: [k=n_local][w], pad 72 (72%4==0)
  const int tid  = threadIdx.x;
  const int wave = tid >> 5, lane = tid & 31, h = lane >> 4, l = lane & 15;
  const int wt = wave & 3, chalf = wave >> 2;
  const int b = blockIdx.y;
  const int nbase = blockIdx.x * 768;
  const float* xb = x + (size_t)b*CH*NPIX;
  v8f acc[8] = {};
  for (int kk0 = 0; kk0 < 768; kk0 += 16) {
    __syncthreads();
    for (int idx = tid; idx < 16*64; idx += 256) {   // 16 rows x 64 float4
      int r = idx >> 6, c4 = (idx & 63) * 4;
      float4 v = *(const float4*)(xb + (size_t)(nbase + kk0 + r)*CH + c4);
      *(float4*)(Xs + r*264 + c4) = v;
    }
    for (int idx = tid; idx < 16*16; idx += 256) {   // 16 rows x 16 float4 of phi
      int r = idx >> 4, w4 = (idx & 15) * 4;
      int n = nbase + kk0 + r;
      float di = dinv[(size_t)b*NPIX + n];
      float4 v = *(const float4*)(phi + (((size_t)b*NPIX + n) << 6) + w4);
      v.x *= di; v.y *= di; v.z *= di; v.w *= di;
      *(float4*)(Ps + r*72 + w4) = v;
    }
    __syncthreads();
    for (int ks = 0; ks < 16; ks += 4) {
      int kk = ks + 2*h;
      v2f a;                                     // A[m=w][k=n]
      a.x = Ps[ kk   *72 + wt*16 + l];
      a.y = Ps[(kk+1)*72 + wt*16 + l];
      #pragma unroll
      for (int ct = 0; ct < 8; ++ct) {
        int c0 = chalf*128 + ct*16;
        v2f bb; bb.x = Xs[kk*264 + c0 + l]; bb.y = Xs[(kk+1)*264 + c0 + l];
        acc[ct] = wmma4(a, bb, acc[ct]);
      }
    }
  }
  #pragma unroll
  for (int ct = 0; ct < 8; ++ct) {
    int c = chalf*128 + ct*16 + l;
    #pragma unroll
    for (int r = 0; r < 8; ++r) {
      int w = wt*16 + r + 8*h;
      tpart[((((size_t)b*12 + blockIdx.x)*WCH + w) << 8) + c] = acc[ct][r];
    }
  }
}

// ---------------- deterministic split-K reduction ----------------
__global__ void k_tred(const float4* __restrict__ tpart, float4* __restrict__ t) {
  int gid = blockIdx.x*256 + threadIdx.x;          // 32768 float4 total
  int b = gid >> 12, rem = gid & 4095;
  float4 s = {0.f, 0.f, 0.f, 0.f};
  for (int j = 0; j < 12; ++j) {
    float4 v = tpart[(((size_t)b*12 + j) << 12) + rem];
    s.x += v.x; s.y += v.y; s.z += v.z; s.w += v.w;
  }
  t[gid] = s;
}

// ---------------- U = -(rou*t) @ Zt   (stored negated) ----------------
__global__ __launch_bounds__(256) void k_U(const float* __restrict__ t,
                                           const float* __restrict__ rou,
                                           const float* __restrict__ zt,
                                           float* __restrict__ Un) {
  __shared__ float As[16*72];
  const int tid  = threadIdx.x;
  const int wave = tid >> 5, lane = tid & 31, h = lane >> 4, l = lane & 15;
  const int wt = wave & 3, chalf = wave >> 2;
  const int b = blockIdx.x;
  v8f acc[8] = {};
  for (int k0 = 0; k0 < CH; k0 += 16) {
    __syncthreads();
    for (int idx = tid; idx < 64*4; idx += 256) {    // 64w x 4 float4 of t rows
      int w = idx >> 2, kq = (idx & 3) * 4;
      float rw_ = rou[b*WCH + w];
      float4 v = *(const float4*)(t + (((size_t)b*WCH + w) << 8) + k0 + kq);
      As[(kq  )*72 + w] = rw_ * v.x;
      As[(kq+1)*72 + w] = rw_ * v.y;
      As[(kq+2)*72 + w] = rw_ * v.z;
      As[(kq+3)*72 + w] = rw_ * v.w;
    }
    __syncthreads();
    for (int ks = 0; ks < 16; ks += 4) {
      int kk = ks + 2*h;
      v2f a; a.x = As[kk*72 + wt*16 + l]; a.y = As[(kk+1)*72 + wt*16 + l];
      #pragma unroll
      for (int ct = 0; ct < 8; ++ct) {
        int c0 = chalf*128 + ct*16;
        v2f bb; bb.x = zt[(k0+kk)*CH + c0 + l]; bb.y = zt[(k0+kk+1)*CH + c0 + l];
        acc[ct] = wmma4(a, bb, acc[ct]);
      }
    }
  }
  #pragma unroll
  for (int ct = 0; ct < 8; ++ct) {
    int c = chalf*128 + ct*16 + l;
    #pragma unroll
    for (int r = 0; r < 8; ++r) {
      int w = wt*16 + r + 8*h;
      Un[(((size_t)b*WCH + w) << 8) + c] = -acc[ct][r];
    }
  }
}

// ---------------- out = x + relu(X@Zt + P@Un + zb) ----------------
__global__ __launch_bounds__(256) void k_final(const float* __restrict__ x,
                                               const float* __restrict__ phi,
                                               const float* __restrict__ dinv,
                                               const float* __restrict__ zt,
                                               const float* __restrict__ Un,
                                               const float* __restrict__ zb,
                                               float* __restrict__ out) {
  __shared__ float sm[64*73];                      // reused by both phases
  const int tid  = threadIdx.x;
  const int wave = tid >> 5, lane = tid & 31, h = lane >> 4, l = lane & 15;
  const int nt = wave & 3, chalf = wave >> 2;
  const int b  = blockIdx.y;
  const int n0 = blockIdx.x * 64;
  const float* xb = x + (size_t)b*CH*NPIX;
  v8f acc[8] = {};
  // phase 1: X @ Zt, K = 256
  for (int k0 = 0; k0 < CH; k0 += 16) {
    __syncthreads();
    for (int idx = tid; idx < 64*4; idx += 256) {  // stage X^T tile: [k][n], b128 reads
      int nl = idx >> 2, kq = (idx & 3) * 4;
      float4 v = *(const float4*)(xb + (size_t)(n0 + nl)*CH + k0 + kq);
      sm[(kq  )*72 + nl] = v.x;
      sm[(kq+1)*72 + nl] = v.y;
      sm[(kq+2)*72 + nl] = v.z;
      sm[(kq+3)*72 + nl] = v.w;
    }
    __syncthreads();
    for (int ks = 0; ks < 16; ks += 4) {
      int kk = ks + 2*h;
      v2f a; a.x = sm[kk*72 + nt*16 + l]; a.y = sm[(kk+1)*72 + nt*16 + l];
      #pragma unroll
      for (int ct = 0; ct < 8; ++ct) {
        int c0 = chalf*128 + ct*16;
        v2f bb; bb.x = zt[(k0+kk)*CH + c0 + l]; bb.y = zt[(k0+kk+1)*CH + c0 + l];
        acc[ct] = wmma4(a, bb, acc[ct]);
      }
    }
  }
  // phase 2: P @ Un, K = 64 (Un pre-negated)
  __syncthreads();
  for (int idx = tid; idx < 64*16; idx += 256) {   // stage P^T: [w][n], pad 73 (gcd(9,64)=1)
    int nl = idx >> 4, w4 = (idx & 15) * 4;
    int n = n0 + nl;
    float di = dinv[(size_t)b*NPIX + n];
    float4 v = *(const float4*)(phi + (((size_t)b*NPIX + n) << 6) + w4);
    sm[(w4  )*73 + nl] = di * v.x;
    sm[(w4+1)*73 + nl] = di * v.y;
    sm[(w4+2)*73 + nl] = di * v.z;
    sm[(w4+3)*73 + nl] = di * v.w;
  }
  __syncthreads();
  for (int ks = 0; ks < 64; ks += 4) {
    int kk = ks + 2*h;
    v2f a; a.x = sm[kk*73 + nt*16 + l]; a.y = sm[(kk+1)*73 + nt*16 + l];
    #pragma unroll
    for (int ct = 0; ct < 8; ++ct) {
      int c0 = chalf*128 + ct*16;
      v2f bb;
      bb.x = Un[(((size_t)b*WCH + kk    ) << 8) + c0 + l];
      bb.y = Un[(((size_t)b*WCH + kk + 1) << 8) + c0 + l];
      acc[ct] = wmma4(a, bb, acc[ct]);
    }
  }
  // epilogue: relu + bias + residual add (flat layout matches reference reshape)
  #pragma unroll
  for (int ct = 0; ct < 8; ++ct) {
    int c = chalf*128 + ct*16 + l;
    float bias = zb[c];
    #pragma unroll
    for (int r = 0; r < 8; ++r) {
      int n = n0 + nt*16 + r + 8*h;
      size_t xi = (size_t)n*CH + c;
      float v = fmaxf(acc[ct][r] + bias, 0.0f);
      out[(size_t)b*CH*NPIX + xi] = xb[xi] + v;
    }
  }
}

extern "C" void kernel_launch(void* const* d_in, const int* in_sizes, int n_in,
                              void* d_out, int out_size, void* d_ws, size_t ws_size,
                              hipStream_t stream) {
  (void)in_sizes; (void)n_in; (void)out_size; (void)ws_size;
  const float* x  = (const float*)d_in[0];
  const float* pw = (const float*)d_in[1];
  const float* pb = (const float*)d_in[2];
  const float* rw = (const float*)d_in[3];
  const float* rb = (const float*)d_in[4];
  const float* zw = (const float*)d_in[5];
  const float* zb = (const float*)d_in[6];
  float* out = (float*)d_out;
  float* ws  = (float*)d_ws;

  float* phi   = ws + OFF_PHI;
  float* dinv  = ws + OFF_DINV;
  float* spart = ws + OFF_SPART;
  float* rs    = ws + OFF_RS;
  float* xmean = ws + OFF_XMEAN;
  float* rou   = ws + OFF_ROU;
  float* tpart = ws + OFF_TPART;
  float* t     = ws + OFF_T;
  float* un    = ws + OFF_U;
  float* zt    = ws + OFF_ZT;

  k_colmean<<<dim3(CH, BATCH), 256, 0, stream>>>(x, xmean);
  k_zt     <<<dim3(CH), CH, 0, stream>>>(zw, zt);
  k_phi    <<<dim3(72, BATCH), 256, 0, stream>>>(x, pw, pb, phi, spart);
  k_rou    <<<dim3(2), 256, 0, stream>>>(xmean, rw, rb, rou);
  k_rs     <<<dim3(2), 256, 0, stream>>>(spart, rou, rs);
  k_dinv   <<<dim3(NPIX/256, BATCH), 256, 0, stream>>>(phi, rs, dinv);
  k_t      <<<dim3(12, BATCH), 256, 0, stream>>>(x, phi, dinv, tpart);
  k_tred   <<<dim3(512), 256, 0, stream>>>((const float4*)tpart, (float4*)t);
  k_U      <<<dim3(BATCH), 256, 0, stream>>>(t, rou, zt, un);
  k_final  <<<dim3(NPIX/64, BATCH), 256, 0, stream>>>(x, phi, dinv, zt, un, zb, out);
}